// GCN_8572754723374
// MI455X (gfx1250) — compile-verified
//
#include <hip/hip_runtime.h>

#define D 128

typedef __attribute__((ext_vector_type(2))) float v2f;
typedef __attribute__((ext_vector_type(8))) float v8f;

// ---------------- degree / normalization ----------------

__global__ void gcn_zero_f32(float* __restrict__ p, int n) {
    int i = blockIdx.x * blockDim.x + threadIdx.x;
    if (i < n) p[i] = 0.0f;
}

__global__ void gcn_count_deg(const int* __restrict__ dst, float* __restrict__ deg, int E) {
    int e = blockIdx.x * blockDim.x + threadIdx.x;
    if (e < E) unsafeAtomicAdd(&deg[dst[e]], 1.0f);
}

__global__ void gcn_make_dinv(float* __restrict__ dinv, int n) {
    int i = blockIdx.x * blockDim.x + threadIdx.x;
    if (i < n) dinv[i] = rsqrtf(dinv[i] + 1.0f);  // +1 for self-loop; always > 0
}

// ---------------- fp32 GEMM via V_WMMA_F32_16X16X4_F32 ----------------
// C[N x 128] = A[N x 128] @ W[128 x 128], row-major.
// One wave computes one 16x16 C tile; block = 8 waves = one 16-row strip
// across all 8 column tiles (A strip reused from cache by all waves).

__global__ void gcn_gemm128(const float* __restrict__ A, const float* __restrict__ W,
                            float* __restrict__ C, int N) {
    const int row0 = blockIdx.x * 16;
    const int wave = threadIdx.x >> 5;
    const int lane = threadIdx.x & 31;
    const int col0 = wave * 16;
    const int m    = lane & 15;   // M (for A-frag) or N (for B-frag / C columns)
    const int kh   = lane >> 4;   // lane-half selects K pair / C row-half

    // clamp row for loads on a ragged tail (N==100000 is a multiple of 16, but be safe)
    int arow_i = row0 + m;
    if (arow_i > N - 1) arow_i = N - 1;
    const float* __restrict__ arow = A + (size_t)arow_i * D;

    v8f acc = {};
#pragma unroll
    for (int k0 = 0; k0 < D; k0 += 4) {
        const int ka = k0 + 2 * kh;  // this lane-half's K base
        // A 16x4 frag: lane m holds A[m][ka], A[m][ka+1]  (contiguous 64-bit load)
        v2f a = *(const v2f*)(arow + ka);
        // B 4x16 frag: lane n holds W[ka][col0+n], W[ka+1][col0+n]
        v2f b;
        b.x = W[(size_t)ka * D + col0 + m];
        b.y = W[(size_t)(ka + 1) * D + col0 + m];
        acc = __builtin_amdgcn_wmma_f32_16x16x4_f32(
            /*neg_a=*/false, a, /*neg_b=*/false, b,
            /*c_mod=*/(short)0, acc, /*reuse_a=*/false, /*reuse_b=*/false);
    }

    // C/D layout: VGPR v -> row (v + 8*kh), column col0 + m
#pragma unroll
    for (int v = 0; v < 8; ++v) {
        int r = row0 + v + 8 * kh;
        if (r < N) C[(size_t)r * D + col0 + m] = acc[v];
    }
}

// ---------------- aggregation ----------------

// out[i,:] = dinv[i]^2 * h[i,:]   (self-loop term; also initializes out)
__global__ void gcn_init_selfloop(const float* __restrict__ h, const float* __restrict__ dinv,
                                  float* __restrict__ out, int nd) {
    int i = blockIdx.x * blockDim.x + threadIdx.x;
    if (i < nd) {
        float dv = dinv[i >> 7];
        out[i] = dv * dv * h[i];
    }
}

// one wave per edge; each lane handles 4 consecutive features (float4 gather + 4 atomics)
__global__ void gcn_edge_scatter(const float* __restrict__ h, const int* __restrict__ src,
                                 const int* __restrict__ dst, const float* __restrict__ dinv,
                                 float* __restrict__ out, int E) {
    long t   = (long)blockIdx.x * blockDim.x + threadIdx.x;
    int  e   = (int)(t >> 5);
    int  ln  = (int)(t & 31);
    if (e >= E) return;
    int s = src[e];
    int d = dst[e];
    float nrm = dinv[s] * dinv[d];
    const float4 v = *((const float4*)(h + (size_t)s * D) + ln);
    float* o = out + (size_t)d * D + ln * 4;
    unsafeAtomicAdd(o + 0, v.x * nrm);
    unsafeAtomicAdd(o + 1, v.y * nrm);
    unsafeAtomicAdd(o + 2, v.z * nrm);
    unsafeAtomicAdd(o + 3, v.w * nrm);
}

__global__ void gcn_bias_act(float* __restrict__ out, const float* __restrict__ b,
                             int nd, int relu) {
    int i = blockIdx.x * blockDim.x + threadIdx.x;
    if (i < nd) {
        float v = out[i] + b[i & (D - 1)];
        out[i] = relu ? fmaxf(v, 0.0f) : v;
    }
}

// ---------------- driver ----------------

extern "C" void kernel_launch(void* const* d_in, const int* in_sizes, int n_in,
                              void* d_out, int out_size, void* d_ws, size_t ws_size,
                              hipStream_t stream) {
    const float* x  = (const float*)d_in[0];
    const float* W0 = (const float*)d_in[1];
    const float* b0 = (const float*)d_in[2];
    const float* W1 = (const float*)d_in[3];
    const float* b1 = (const float*)d_in[4];
    const float* W2 = (const float*)d_in[5];
    const float* b2 = (const float*)d_in[6];
    const int*   ei = (const int*)d_in[7];

    const int N = in_sizes[0] / D;
    const int E = in_sizes[7] / 2;
    const int* src = ei;
    const int* dst = ei + E;

    float* out = (float*)d_out;

    char*  ws   = (char*)d_ws;
    float* dinv = (float*)ws;
    size_t off  = ((size_t)N * sizeof(float) + 255) & ~(size_t)255;
    float* bufA = (float*)(ws + off);
    float* bufB = bufA + (size_t)N * D;

    const dim3 blk(256);
    const int  nd       = N * D;
    const int  nBlkN    = (N + 255) / 256;
    const int  nBlkE    = (E + 255) / 256;
    const int  nBlkND   = (nd + 255) / 256;
    const int  nBlkGemm = (N + 15) / 16;
    const long sthreads = (long)E * 32;
    const int  nBlkScat = (int)((sthreads + 255) / 256);

    // normalization: deg -> dinv
    gcn_zero_f32<<<nBlkN, blk, 0, stream>>>(dinv, N);
    gcn_count_deg<<<nBlkE, blk, 0, stream>>>(dst, dinv, E);
    gcn_make_dinv<<<nBlkN, blk, 0, stream>>>(dinv, N);

    // layer 0: x -> bufB (ReLU)
    gcn_gemm128<<<nBlkGemm, blk, 0, stream>>>(x, W0, bufA, N);
    gcn_init_selfloop<<<nBlkND, blk, 0, stream>>>(bufA, dinv, bufB, nd);
    gcn_edge_scatter<<<nBlkScat, blk, 0, stream>>>(bufA, src, dst, dinv, bufB, E);
    gcn_bias_act<<<nBlkND, blk, 0, stream>>>(bufB, b0, nd, 1);

    // layer 1: bufB -> bufB (ReLU), via bufA
    gcn_gemm128<<<nBlkGemm, blk, 0, stream>>>(bufB, W1, bufA, N);
    gcn_init_selfloop<<<nBlkND, blk, 0, stream>>>(bufA, dinv, bufB, nd);
    gcn_edge_scatter<<<nBlkScat, blk, 0, stream>>>(bufA, src, dst, dinv, bufB, E);
    gcn_bias_act<<<nBlkND, blk, 0, stream>>>(bufB, b1, nd, 1);

    // layer 2: bufB -> d_out (no ReLU)
    gcn_gemm128<<<nBlkGemm, blk, 0, stream>>>(bufB, W2, bufA, N);
    gcn_init_selfloop<<<nBlkND, blk, 0, stream>>>(bufA, dinv, out, nd);
    gcn_edge_scatter<<<nBlkScat, blk, 0, stream>>>(bufA, src, dst, dinv, out, E);
    gcn_bias_act<<<nBlkND, blk, 0, stream>>>(out, b2, nd, 0);
}